// VocabParallelEmbeddingWithLoRA_16157666967997
// MI455X (gfx1250) — compile-verified
//
#include <hip/hip_runtime.h>

typedef __attribute__((ext_vector_type(2))) float v2f;
typedef __attribute__((ext_vector_type(8))) float v8f;

#define ORG_VOCAB   128000
#define EXTRA_VOCAB 256
#define FULL_VOCAB  128256
#define EMBED_DIM   2048
#define RANK        16
#define MAX_LORAS   8
#define KDIM        (MAX_LORAS * RANK)   // 128
#define TOK_TILE    16
#define BLOCK_THREADS 256
#define WAVES_PER_BLOCK (BLOCK_THREADS / 32)
#define COLS_PER_WAVE   (EMBED_DIM / WAVES_PER_BLOCK)   // 256
#define NTILES_PER_WAVE (COLS_PER_WAVE / 16)            // 16

__global__ __launch_bounds__(BLOCK_THREADS)
void lora_embed_wmma_kernel(const int* __restrict__ x,
                            const int* __restrict__ lora_idx,
                            const float* __restrict__ weight,
                            const float* __restrict__ lora_a,
                            const float* __restrict__ lora_b,
                            float* __restrict__ out,
                            int num_tokens)
{
    // One-hot-expanded A tile: 16 tokens x 128 (= 8 loras * rank 16), f32.
    __shared__ float Asm[TOK_TILE][KDIM];    // 8 KB
    __shared__ int   baseRow[TOK_TILE];

    const int tid    = threadIdx.x;
    const int block0 = blockIdx.x * TOK_TILE;

    // --- Phase 1: build A_ext in LDS + base-row indices --------------------
    #pragma unroll
    for (int i = 0; i < (TOK_TILE * KDIM) / BLOCK_THREADS; ++i)
        ((float*)Asm)[tid + i * BLOCK_THREADS] = 0.0f;
    __syncthreads();

    {
        const int m = tid >> 4;            // token within tile, 0..15
        const int r = tid & 15;            // rank element,      0..15
        int tIdx = block0 + m;
        if (tIdx >= num_tokens) tIdx = num_tokens - 1;   // safe clamp
        const int tok = x[tIdx];
        const int l   = lora_idx[tIdx];
        const float av = lora_a[((size_t)l * FULL_VOCAB + tok) * RANK + r];
        Asm[m][l * RANK + r] = av;         // one-hot K-slot for this lora
        if (r == 0) {
            baseRow[m] = tok + ((tok > ORG_VOCAB - 1) ? l * EXTRA_VOCAB : 0);
        }
    }
    __syncthreads();

    // --- Phase 2: per-wave GEMM delta + fused gather/add/store -------------
    const int lane = tid & 31;
    const int wave = tid >> 5;
    const int half = lane >> 4;            // 0: K offset 0,1 ; 1: K offset 2,3
    const int nl   = lane & 15;            // A-row (M) / B-col (N) within tile

    // Preload all 32 K-step A fragments (v2f each) from LDS.
    // A 16x4 f32 layout: M = lane&15 for both halves; VGPR0=K0(+2*half), VGPR1=K0+1.
    v2f afrag[32];
    #pragma unroll
    for (int s = 0; s < 32; ++s) {
        const int k0 = 4 * s + 2 * half;
        const float2 t = *(const float2*)&Asm[nl][k0];   // k0 even -> 8B aligned
        afrag[s].x = t.x; afrag[s].y = t.y;
    }

    const int colBase = wave * COLS_PER_WAVE;

    #pragma unroll 1
    for (int j = 0; j < NTILES_PER_WAVE; ++j) {
        const int col = colBase + j * 16 + nl;
        v8f c = {};
        // delta(16x16) = sum over K=128 of A_ext * B_cat^T, 32 x WMMA f32 16x16x4
        #pragma unroll
        for (int s = 0; s < 32; ++s) {
            const int k0 = 4 * s + 2 * half;
            const int l  = k0 >> 4;        // lora slot
            const int r  = k0 & 15;        // rank offset (even -> float2 aligned)
            const float2 tb =
                *(const float2*)&lora_b[((size_t)l * EMBED_DIM + col) * RANK + r];
            v2f bfrag; bfrag.x = tb.x; bfrag.y = tb.y;
            c = __builtin_amdgcn_wmma_f32_16x16x4_f32(
                    /*neg_a=*/false, afrag[s],
                    /*neg_b=*/false, bfrag,
                    /*c_mod=*/(short)0, c,
                    /*reuse_a=*/false, /*reuse_b=*/false);
        }
        // Epilogue: C layout -> VGPR i holds (M = i + 8*half, N = nl).
        #pragma unroll
        for (int i = 0; i < 8; ++i) {
            const int m = i + half * 8;
            const int t = block0 + m;
            if (t < num_tokens) {
                const float w = weight[(size_t)baseRow[m] * EMBED_DIM + col];
                out[(size_t)t * EMBED_DIM + col] = w + c[i];
            }
        }
    }
}

extern "C" void kernel_launch(void* const* d_in, const int* in_sizes, int n_in,
                              void* d_out, int out_size, void* d_ws, size_t ws_size,
                              hipStream_t stream) {
    const int*   x        = (const int*)d_in[0];
    const int*   lora_idx = (const int*)d_in[1];
    const float* weight   = (const float*)d_in[2];
    const float* lora_a   = (const float*)d_in[3];
    const float* lora_b   = (const float*)d_in[4];
    float*       out      = (float*)d_out;

    const int num_tokens = in_sizes[0];
    const int grid = (num_tokens + TOK_TILE - 1) / TOK_TILE;   // 512 for T=8192

    lora_embed_wmma_kernel<<<grid, BLOCK_THREADS, 0, stream>>>(
        x, lora_idx, weight, lora_a, lora_b, out, num_tokens);
}